// Template2Pair_65377992179723
// MI455X (gfx1250) — compile-verified
//
#include <hip/hip_runtime.h>
#include <hip/hip_bf16.h>

// ---- problem constants (B=1) ----
#define LSEQ   384
#define TILE   16
#define NQ1    256
#define NQ2    128     // pair feature dim == KEY_DIM
#define NTI    88      // templ input dim
#define NIN    64      // n_inner_emb
#define HALFD  32
#define KEYD   128
#define TN     4
#define EPS    1e-5f

// padded LDS row strides (bank-conflict free: 64 banks x 4B)
#define LDA    132     // for 128-wide tiles: bank = (4m + k) % 64, all lanes distinct
#define LDN    68      // for 64-wide tiles:  bank = (4r + c) % 64, all lanes distinct

typedef float v2f __attribute__((ext_vector_type(2)));
typedef float v8f __attribute__((ext_vector_type(8)));

__device__ __forceinline__ float wave_reduce_add(float v) {
#pragma unroll
  for (int off = 16; off > 0; off >>= 1) v += __shfl_xor(v, off, 32);
  return v;
}

// ---------------------------------------------------------------------------
// Kernel 1: left/right = relu(LN(seq @ W)), one wave per output row.
// wave 0 -> left, wave 1 -> right; lane = output channel (32 == wave32).
// ---------------------------------------------------------------------------
__global__ __launch_bounds__(64) void seq_embed_kernel(
    const float* __restrict__ seq,
    const float* __restrict__ Wl, const float* __restrict__ gl, const float* __restrict__ bl,
    const float* __restrict__ Wr, const float* __restrict__ gr, const float* __restrict__ br,
    float* __restrict__ leftArr, float* __restrict__ rightArr) {
  const int row  = blockIdx.x;
  const int wave = threadIdx.x >> 5;
  const int lane = threadIdx.x & 31;
  const float* W = wave ? Wr : Wl;
  const float* g = wave ? gr : gl;
  const float* b = wave ? br : bl;
  float* out     = wave ? rightArr : leftArr;
  const float* s = seq + (size_t)row * NQ1;
  float acc = 0.f;
#pragma unroll 4
  for (int k = 0; k < NQ1; ++k) acc = fmaf(s[k], W[k * HALFD + lane], acc);
  const float mu  = wave_reduce_add(acc) * (1.f / HALFD);
  const float d   = acc - mu;
  const float var = wave_reduce_add(d * d) * (1.f / HALFD);
  const float v   = d * rsqrtf(var + EPS) * g[lane] + b[lane];
  out[(size_t)row * HALFD + lane] = fmaxf(v, 0.f);
}

// ---------------------------------------------------------------------------
// WMMA helpers: f32 16x16x4, A (16xK) in LDS with padded row stride lda,
// B (KxN=64) row-major in global (L2-resident weights).
// A layout (ISA 7.12.2): lanes 0-15 M=lane {K=kb,kb+1}, lanes 16-31 {K=kb+2,kb+3}.
// B layout mirrors C row-striping: VGPR idx -> K row, hi lanes -> K+2.
// C/D: VGPR r, col = lane&15, row = r + 8*(lane>=16).
// ---------------------------------------------------------------------------
__device__ __forceinline__ v8f gemm16(const float* __restrict__ A, int lda,
                                      const float* __restrict__ B,   // KxN, ldb = 64
                                      int K, int ncol0, int lane) {
  const int m   = lane & 15;
  const int klo = (lane >> 4) << 1;           // 0 or 2
  const int n   = ncol0 + m;
  v8f c = {};
  for (int kb = 0; kb < K; kb += 4) {
    v2f a, b;
    a.x = A[m * lda + kb + klo];              // 8B-aligned pair -> ds_load b64
    a.y = A[m * lda + kb + klo + 1];
    b.x = B[(kb + klo) * NIN + n];
    b.y = B[(kb + klo + 1) * NIN + n];
    c = __builtin_amdgcn_wmma_f32_16x16x4_f32(false, a, false, b, (short)0, c,
                                              false, false);
  }
  return c;
}

__device__ __forceinline__ void store_c(float* __restrict__ out, int ldo,
                                        int ncol0, int lane, v8f c) {
  const int hi = lane >> 4;
  const int n  = ncol0 + (lane & 15);
#pragma unroll
  for (int r = 0; r < 8; ++r) out[(r + hi * 8) * ldo + n] = c[r];
}

// LayerNorm over C columns for 16 rows; threads 0..15 each own one row.
// in/out may alias. Padded strides keep the 16 row-walkers on distinct banks.
__device__ __forceinline__ void ln_rows(const float* __restrict__ in, int ldi,
                                        float* __restrict__ out, int ldo, int C,
                                        const float* __restrict__ g,
                                        const float* __restrict__ b,
                                        bool relu, int tid) {
  if (tid < TILE) {
    const float* r = in + tid * ldi;
    float mu = 0.f;
    for (int c = 0; c < C; ++c) mu += r[c];
    mu /= (float)C;
    float var = 0.f;
    for (int c = 0; c < C; ++c) { float d = r[c] - mu; var = fmaf(d, d, var); }
    var /= (float)C;
    const float rs = rsqrtf(var + EPS);
    float* o = out + tid * ldo;
    for (int c = 0; c < C; ++c) {
      float v = (r[c] - mu) * rs * g[c] + b[c];
      o[c] = relu ? fmaxf(v, 0.f) : v;
    }
  }
}

// ---------------------------------------------------------------------------
// Kernel 2: per 16-row (i, j0..j0+15) tile: pair proj -> z -> q -> template
// proj -> softmax attention -> final LN. 128 threads = 4 wave32.
// ---------------------------------------------------------------------------
__global__ __launch_bounds__(128) void t2p_kernel(
    const float* __restrict__ pair_feat, const float* __restrict__ templ_feat,
    const float* __restrict__ templ_mask,
    const float* __restrict__ Wp, const float* __restrict__ gp, const float* __restrict__ bp,
    const float* __restrict__ Wt, const float* __restrict__ gt, const float* __restrict__ bt,
    const float* __restrict__ gnt, const float* __restrict__ bnt,
    const float* __restrict__ Wq,
    const float* __restrict__ gout, const float* __restrict__ bout,
    const float* __restrict__ leftArr, const float* __restrict__ rightArr,
    float* __restrict__ out) {
  const int tb   = blockIdx.x;
  const int r0   = tb * TILE;        // flattened (i*L + j) base
  const int i    = r0 / LSEQ;
  const int j0   = r0 % LSEQ;
  const int tid  = threadIdx.x;
  const int lane = tid & 31;
  const int wave = tid >> 5;
  const int nc0  = wave * 16;        // this wave's output-column tile

  __shared__ float sA[TILE][LDA];       // staging: pair 16x128 / templ 16x88
  __shared__ float sZ[TILE][LDA];       // z then zn (128 cols used)
  __shared__ float sRaw[TILE][LDN];     // raw GEMM outputs / attention out
  __shared__ float sQ[TILE][LDN];
  __shared__ float sT[TN][TILE][LDN];
  __shared__ float sAttn[TILE][TN];
  __shared__ float sStat[TILE][2];
  __shared__ float sMask[TN];

  if (tid < TN) sMask[tid] = templ_mask[tid];

  // ---- Phase 1: pair = relu(LN(pair_feat @ Wp)) -> sZ[:,64:128] ----
  for (int idx = tid; idx < TILE * NQ2; idx += 128) {
    const int r = idx >> 7, c = idx & 127;
    sA[r][c] = pair_feat[(size_t)(r0 + r) * NQ2 + c];
  }
  __syncthreads();
  {
    v8f c = gemm16(&sA[0][0], LDA, Wp, NQ2, nc0, lane);
    store_c(&sRaw[0][0], LDN, nc0, lane, c);
  }
  __syncthreads();
  ln_rows(&sRaw[0][0], LDN, &sZ[0][2 * HALFD], LDA, NIN, gp, bp, true, tid);
  // z[:,0:32] = left[i], z[:,32:64] = right[j0+r]
  for (int idx = tid; idx < TILE * 2 * HALFD; idx += 128) {
    const int r = idx >> 6, c = idx & 63;
    sZ[r][c] = (c < HALFD) ? leftArr[(size_t)i * HALFD + c]
                           : rightArr[(size_t)(j0 + r) * HALFD + (c - HALFD)];
  }
  __syncthreads();
  // ---- Phase 2: zn = LN(z) over 128 (in place) ----
  ln_rows(&sZ[0][0], LDA, &sZ[0][0], LDA, KEYD, gnt, bnt, false, tid);
  __syncthreads();
  // ---- Phase 3: q = zn @ Wq ----
  {
    v8f c = gemm16(&sZ[0][0], LDA, Wq, KEYD, nc0, lane);
    store_c(&sQ[0][0], LDN, nc0, lane, c);
  }
  // ---- Phase 4: templates t = relu(LN(templ_feat @ Wt)) ----
  for (int t = 0; t < TN; ++t) {
    __syncthreads();   // sA / sRaw free for reuse
    for (int idx = tid; idx < TILE * NTI; idx += 128) {
      const int r = idx / NTI, c = idx - r * NTI;
      sA[r][c] = templ_feat[((size_t)(t * LSEQ + i) * LSEQ + j0 + r) * NTI + c];
    }
    __syncthreads();
    v8f c = gemm16(&sA[0][0], LDA, Wt, NTI, nc0, lane);
    store_c(&sRaw[0][0], LDN, nc0, lane, c);
    __syncthreads();
    ln_rows(&sRaw[0][0], LDN, &sT[t][0][0], LDN, NIN, gt, bt, true, tid);
  }
  __syncthreads();

  // ---- Phase 5: scores (row,t) on 64 threads, softmax on 16 threads ----
  if (tid < TILE * TN) {
    const int r = tid >> 2, t = tid & 3;
    float s = 0.f;
#pragma unroll 4
    for (int c = 0; c < NIN; ++c) s = fmaf(sQ[r][c], sT[t][r][c], s);
    s *= 0.125f;                                   // 1/sqrt(64)
    if (!(sMask[t] > 0.f)) s -= 1e9f;
    sAttn[r][t] = s;
  }
  __syncthreads();
  if (tid < TILE) {
    float m = sAttn[tid][0];
#pragma unroll
    for (int t = 1; t < TN; ++t) m = fmaxf(m, sAttn[tid][t]);
    float e[TN], sum = 0.f;
#pragma unroll
    for (int t = 0; t < TN; ++t) { e[t] = expf(sAttn[tid][t] - m); sum += e[t]; }
    const float inv = 1.f / sum;
#pragma unroll
    for (int t = 0; t < TN; ++t) sAttn[tid][t] = e[t] * inv;
  }
  __syncthreads();
  // ---- attention-weighted sum over templates ----
  for (int idx = tid; idx < TILE * NIN; idx += 128) {
    const int r = idx >> 6, c = idx & 63;
    float v = 0.f;
#pragma unroll
    for (int t = 0; t < TN; ++t) v = fmaf(sAttn[r][t], sT[t][r][c], v);
    sRaw[r][c] = v;
  }
  __syncthreads();
  // ---- final LN over 64, coalesced store ----
  if (tid < TILE) {
    float mu = 0.f;
    for (int c = 0; c < NIN; ++c) mu += sRaw[tid][c];
    mu *= (1.f / NIN);
    float var = 0.f;
    for (int c = 0; c < NIN; ++c) { float d = sRaw[tid][c] - mu; var = fmaf(d, d, var); }
    var *= (1.f / NIN);
    sStat[tid][0] = mu;
    sStat[tid][1] = rsqrtf(var + EPS);
  }
  __syncthreads();
  for (int idx = tid; idx < TILE * NIN; idx += 128) {
    const int r = idx >> 6, c = idx & 63;
    out[(size_t)(r0 + r) * NIN + c] =
        (sRaw[r][c] - sStat[r][0]) * sStat[r][1] * gout[c] + bout[c];
  }
}

// ---------------------------------------------------------------------------
extern "C" void kernel_launch(void* const* d_in, const int* in_sizes, int n_in,
                              void* d_out, int out_size, void* d_ws, size_t ws_size,
                              hipStream_t stream) {
  const float* seq   = (const float*)d_in[0];
  const float* pairf = (const float*)d_in[1];
  const float* templ = (const float*)d_in[2];
  const float* tmask = (const float*)d_in[3];
  const float* Wl  = (const float*)d_in[4];
  const float* gl  = (const float*)d_in[5];
  const float* bl  = (const float*)d_in[6];
  const float* Wr  = (const float*)d_in[7];
  const float* gr  = (const float*)d_in[8];
  const float* br  = (const float*)d_in[9];
  const float* Wp  = (const float*)d_in[10];
  const float* gp  = (const float*)d_in[11];
  const float* bp  = (const float*)d_in[12];
  const float* Wt  = (const float*)d_in[13];
  const float* gt  = (const float*)d_in[14];
  const float* bt  = (const float*)d_in[15];
  const float* gnt = (const float*)d_in[16];
  const float* bnt = (const float*)d_in[17];
  const float* Wq  = (const float*)d_in[18];
  const float* go  = (const float*)d_in[19];
  const float* bo  = (const float*)d_in[20];

  float* leftArr  = (float*)d_ws;                   // 384*32
  float* rightArr = leftArr + (size_t)LSEQ * HALFD; // 384*32
  float* outp     = (float*)d_out;

  seq_embed_kernel<<<LSEQ, 64, 0, stream>>>(seq, Wl, gl, bl, Wr, gr, br,
                                            leftArr, rightArr);

  const int nTiles = (LSEQ * LSEQ) / TILE;          // 9216
  t2p_kernel<<<nTiles, 128, 0, stream>>>(pairf, templ, tmask,
                                         Wp, gp, bp, Wt, gt, bt,
                                         gnt, bnt, Wq, go, bo,
                                         leftArr, rightArr, outp);
}